// HebbianLayer_18640158065317
// MI455X (gfx1250) — compile-verified
//
#include <hip/hip_runtime.h>
#include <hip/hip_bf16.h>
#include <math.h>

typedef __attribute__((ext_vector_type(2))) float v2f;
typedef __attribute__((ext_vector_type(8))) float v8f;

#define ALPHA_SCALE 9.9f
#define EPS_LN 1e-5f
#define EPS_NORM 1e-12f

// ---------------------------------------------------------------------------
// Block-level GEMM: Y[32,N] = act(X[32,K] @ W[K,N] + bias) for ONE 16-col
// strip per block. blockDim=256 (8 waves): K reduction split across the 8
// waves (deterministic LDS combine). K,N are compile-time so all strides
// fold into immediate offsets; inner loop is pointer-increment only.
// ---------------------------------------------------------------------------
template <int K, int N>
__device__ __forceinline__ void gemm32_core(const float* __restrict__ X,
                                            const float* __restrict__ W,
                                            const float* __restrict__ bias,
                                            float* __restrict__ Y,
                                            int n0, int do_relu,
                                            float (*red)[32][17]) {
    constexpr int Kc = K >> 3;             // per-wave K chunk
    constexpr int PF = 64;                 // prefetch distance (k rows)
    constexpr int mainK = (Kc > PF) ? (Kc - PF) : 0;

    const int tid  = threadIdx.x;
    const int lane = tid & 31;
    const int wave = tid >> 5;             // 0..7
    const int half = lane >> 4;
    const int l15  = lane & 15;
    const int col  = n0 + l15;
    const int k0   = wave * Kc;

    const float* pA0 = X + (size_t)l15        * K + k0 + half * 2;
    const float* pA1 = X + (size_t)(l15 + 16) * K + k0 + half * 2;
    const float* pB  = W + (size_t)(k0 + half * 2) * N + col;

    v8f c0 = {};                           // rows 0..15 partial
    v8f c1 = {};                           // rows 16..31 partial
    int k = 0;
    for (; k < mainK; k += 4) {            // main body: constant-offset prefetch
        __builtin_prefetch(pB + (size_t)PF * N, 0, 3);
        v2f a0, a1, b;
        a0.x = pA0[0]; a0.y = pA0[1];
        a1.x = pA1[0]; a1.y = pA1[1];
        b.x  = pB[0];  b.y  = pB[N];
        c0 = __builtin_amdgcn_wmma_f32_16x16x4_f32(false, a0, false, b, (short)0, c0, false, false);
        c1 = __builtin_amdgcn_wmma_f32_16x16x4_f32(false, a1, false, b, (short)0, c1, false, false);
        pA0 += 4; pA1 += 4; pB += (size_t)4 * N;
    }
    for (; k < Kc; k += 4) {               // tail: no prefetch
        v2f a0, a1, b;
        a0.x = pA0[0]; a0.y = pA0[1];
        a1.x = pA1[0]; a1.y = pA1[1];
        b.x  = pB[0];  b.y  = pB[N];
        c0 = __builtin_amdgcn_wmma_f32_16x16x4_f32(false, a0, false, b, (short)0, c0, false, false);
        c1 = __builtin_amdgcn_wmma_f32_16x16x4_f32(false, a1, false, b, (short)0, c1, false, false);
        pA0 += 4; pA1 += 4; pB += (size_t)4 * N;
    }
#pragma unroll
    for (int r = 0; r < 8; ++r) {
        red[wave][lane][r]     = c0[r];
        red[wave][lane][8 + r] = c1[r];
    }
    __syncthreads();
    // 512 outputs, 256 threads -> each finalizes 2 (fixed-order sums)
    const int rr = tid >> 5;               // 0..7
    float s0 = 0.0f, s1 = 0.0f;
#pragma unroll
    for (int w = 0; w < 8; ++w) {
        s0 += red[w][lane][rr];
        s1 += red[w][lane][rr + 8];
    }
    const float bv = bias ? bias[col] : 0.0f;
    s0 += bv; s1 += bv;
    if (do_relu) { s0 = fmaxf(s0, 0.0f); s1 = fmaxf(s1, 0.0f); }
    const int row0 = rr + 8 * half;        // D-tile layout: VGPR r -> M=r (+8 hi half)
    Y[(size_t)row0        * N + col] = s0;
    Y[(size_t)(row0 + 16) * N + col] = s1;
}

// both big forward GEMMs in one launch: grid=(D/16, 2)
__global__ void fwd_dual_gemm(const float* __restrict__ stimulus,
                              const float* __restrict__ W_heb,
                              const float* __restrict__ prev_act,
                              const float* __restrict__ W_rec,
                              float* __restrict__ stim_out,
                              float* __restrict__ rec_out) {
    __shared__ float red[8][32][17];
    const float* X = blockIdx.y ? prev_act : stimulus;
    const float* W = blockIdx.y ? W_rec    : W_heb;
    float*       Y = blockIdx.y ? rec_out  : stim_out;
    gemm32_core<4096, 4096>(X, W, nullptr, Y, blockIdx.x * 16, 1, red);
}

// templated single-matrix version for the MLP layers: grid=(N/16)
template <int K, int N>
__global__ void gemm32_block(const float* __restrict__ X,
                             const float* __restrict__ W,
                             const float* __restrict__ bias,
                             float* __restrict__ Y, int do_relu) {
    __shared__ float red[8][32][17];
    gemm32_core<K, N>(X, W, bias, Y, blockIdx.x * 16, do_relu, red);
}

// ---------------------------------------------------------------------------
// final[b,:] = stim_out[b,:] + LN(rec_out[b,:])*g + bb ; also writes d_out row b
// ---------------------------------------------------------------------------
__device__ __forceinline__ float block_reduce_256(float v, float* red) {
    red[threadIdx.x] = v;
    __syncthreads();
    for (int off = 128; off > 0; off >>= 1) {
        if ((int)threadIdx.x < off) red[threadIdx.x] += red[threadIdx.x + off];
        __syncthreads();
    }
    float r = red[0];
    __syncthreads();
    return r;
}

__global__ void ln_rec_add(const float* __restrict__ rec,
                           const float* __restrict__ stim,
                           const float* __restrict__ g,
                           const float* __restrict__ bb,
                           float* __restrict__ final_ws,
                           float* __restrict__ out, int D) {
    __shared__ float red[256];
    const int b = blockIdx.x;
    float s = 0.0f;
    for (int j = threadIdx.x; j < D; j += 256) s += rec[(size_t)b * D + j];
    const float mean = block_reduce_256(s, red) / (float)D;
    float v = 0.0f;
    for (int j = threadIdx.x; j < D; j += 256) {
        float d = rec[(size_t)b * D + j] - mean;
        v += d * d;
    }
    const float var = block_reduce_256(v, red) / (float)D;
    const float inv = rsqrtf(var + EPS_LN);
    for (int j = threadIdx.x; j < D; j += 256) {
        float r = (rec[(size_t)b * D + j] - mean) * inv * g[j] + bb[j];
        float f = stim[(size_t)b * D + j] + r;
        final_ws[(size_t)b * D + j] = f;
        out[(size_t)b * D + j] = f;
    }
}

__global__ void ln_relu_rows(const float* __restrict__ X,
                             const float* __restrict__ g,
                             const float* __restrict__ bb,
                             float* __restrict__ Y, int L) {
    __shared__ float red[256];
    const int row = blockIdx.x;
    float s = 0.0f;
    for (int j = threadIdx.x; j < L; j += 256) s += X[(size_t)row * L + j];
    const float mean = block_reduce_256(s, red) / (float)L;
    float v = 0.0f;
    for (int j = threadIdx.x; j < L; j += 256) {
        float d = X[(size_t)row * L + j] - mean;
        v += d * d;
    }
    const float var = block_reduce_256(v, red) / (float)L;
    const float inv = rsqrtf(var + EPS_LN);
    for (int j = threadIdx.x; j < L; j += 256) {
        float y = (X[(size_t)row * L + j] - mean) * inv * g[j] + bb[j];
        Y[(size_t)row * L + j] = fmaxf(y, 0.0f);
    }
}

// ---------------------------------------------------------------------------
// dop = mean_b tanh(LN(h3)*g+b);  alpha[i] = dop[i/64]*9.9
// one block, 1024 threads (32 waves, wave b = batch row b). Deterministic.
// ---------------------------------------------------------------------------
__global__ void mlp_tail(const float* __restrict__ h3,
                         const float* __restrict__ g,
                         const float* __restrict__ bb,
                         float* __restrict__ alpha, int D) {
    __shared__ float tv[32][64];
    __shared__ float dop[64];
    const int lane = threadIdx.x & 31;
    const int wave = threadIdx.x >> 5;     // batch row
    const float x0 = h3[wave * 64 + lane];
    const float x1 = h3[wave * 64 + lane + 32];
    float s = x0 + x1;
    for (int off = 16; off > 0; off >>= 1) s += __shfl_xor(s, off, 32);
    const float mean = s * (1.0f / 64.0f);
    const float d0 = x0 - mean, d1 = x1 - mean;
    float v = d0 * d0 + d1 * d1;
    for (int off = 16; off > 0; off >>= 1) v += __shfl_xor(v, off, 32);
    const float inv = rsqrtf(v * (1.0f / 64.0f) + EPS_LN);
    tv[wave][lane]      = tanhf(d0 * inv * g[lane]      + bb[lane]);
    tv[wave][lane + 32] = tanhf(d1 * inv * g[lane + 32] + bb[lane + 32]);
    __syncthreads();
    if (threadIdx.x < 64) {
        float acc = 0.0f;
        for (int b = 0; b < 32; ++b) acc += tv[b][threadIdx.x];
        dop[threadIdx.x] = acc * (1.0f / 32.0f);
    }
    __syncthreads();
    for (int i = threadIdx.x; i < D; i += blockDim.x)
        alpha[i] = dop[i >> 6] * ALPHA_SCALE;
}

// ---------------------------------------------------------------------------
// Update phase, both matrices in one launch (grid=(D/16, 2)):
//   T[i,j] = (W[i,j] + (sum_b X[b,i]*Yact[b,j]) * alpha[j]) * (1-decay[i])
//   out[i,:] = T[i,:] / max(||T[i,:]||_2, 1e-12)
// One workgroup per 16-row panel; the full 16x4096 f32 panel lives in 256 KB
// of dynamic LDS (WGP has 320 KB) so W is read ONCE and written ONCE.
// D is compile-time (4096): all row strides are immediate offsets off
// base pointers that advance by a constant 128 floats per j-tile step.
// ---------------------------------------------------------------------------
__global__ void update_dual(const float* __restrict__ W_heb,
                            const float* __restrict__ W_rec,
                            const float* __restrict__ stimulus,
                            const float* __restrict__ prev_act,
                            const float* __restrict__ stim_out,
                            const float* __restrict__ rec_out,
                            const float* __restrict__ alpha,
                            const float* __restrict__ decay,
                            float* __restrict__ out1,
                            float* __restrict__ out2) {
    constexpr int D  = 4096;
    constexpr int NT = D / 16;             // 256 j-tiles
    extern __shared__ float ldsT[];        // 16 * D floats (dynamic, 256 KB)
    __shared__ float redsq[8][32][8];
    __shared__ float rscale[16];

    const float* W    = blockIdx.y ? W_rec    : W_heb;
    const float* Xin  = blockIdx.y ? prev_act : stimulus;
    const float* Yact = blockIdx.y ? rec_out  : stim_out;
    float*       out  = blockIdx.y ? out2     : out1;

    const int lane = threadIdx.x & 31;
    const int wave = threadIdx.x >> 5;     // 0..7
    const int i0   = blockIdx.x * 16;      // row panel
    const int half = lane >> 4;
    const int l15  = lane & 15;

    // A operand: A[M=i, K=b] = Xin[b][i0+M] ; independent of j -> load once.
    v2f a[8];
#pragma unroll
    for (int ks = 0; ks < 8; ++ks) {
        const int b0 = ks * 4 + half * 2;
        a[ks].x = Xin[(size_t)(b0 + 0) * D + i0 + l15];
        a[ks].y = Xin[(size_t)(b0 + 1) * D + i0 + l15];
    }
    float decv[8];
#pragma unroll
    for (int r = 0; r < 8; ++r) decv[r] = 1.0f - decay[i0 + r + 8 * half];

    float acc[8];
#pragma unroll
    for (int r = 0; r < 8; ++r) acc[r] = 0.0f;

    // per-iteration state: column base advances by 128 (8 waves * 16 cols)
    const int col0 = wave * 16 + l15;
    const float* pY  = Yact + col0;                       // + b*D immediates
    const float* pW  = W + (size_t)i0 * D + col0;         // + row*D immediates
    const float* pAl = alpha + col0;
    const float* pWp = W + (size_t)(i0 + l15) * D + wave * 16 + 1024; // prefetch: row l15, 8 tiles ahead
    int ldsIdx = col0;

    constexpr int ITER  = NT / 8;          // 32 j-tiles per wave
    constexpr int MAINI = ITER - 8;        // last 8 iterations: no prefetch

#pragma unroll 1
    for (int it = 0; it < ITER; ++it) {
        if (it < MAINI) __builtin_prefetch(pWp, 0, 3);
        v8f c = {};
#pragma unroll
        for (int ks = 0; ks < 8; ++ks) {
            const int b0 = ks * 4 + half * 2;
            v2f b;
            b.x = pY[(size_t)(b0 + 0) * D];
            b.y = pY[(size_t)(b0 + 1) * D];
            c = __builtin_amdgcn_wmma_f32_16x16x4_f32(false, a[ks], false, b, (short)0, c, false, false);
        }
        const float al = pAl[0];
#pragma unroll
        for (int r = 0; r < 8; ++r) {
            const int row = r + 8 * half;
            const float t = (pW[(size_t)row * D] + c[r] * al) * decv[r];
            ldsT[row * D + ldsIdx] = t;
            acc[r] += t * t;
        }
        pY += 128; pW += 128; pAl += 128; pWp += 128; ldsIdx += 128;
    }
#pragma unroll
    for (int r = 0; r < 8; ++r) redsq[wave][lane][r] = acc[r];
    __syncthreads();

    // deterministic row-norm: thread t (<16) sums its row's 128 partials in order
    if (threadIdx.x < 16) {
        const int row = threadIdx.x;
        const int hb  = (row >> 3) * 16;   // rows 0-7 in lanes 0-15, 8-15 in 16-31
        const int r   = row & 7;
        float s = 0.0f;
        for (int w = 0; w < 8; ++w)
            for (int l = 0; l < 16; ++l)
                s += redsq[w][hb + l][r];
        rscale[row] = 1.0f / fmaxf(sqrtf(s), EPS_NORM);
    }
    __syncthreads();

    // stream the normalized panel out of LDS
#pragma unroll 1
    for (int row = 0; row < 16; ++row) {
        const float sc = rscale[row];
        float* orow = out + (size_t)(i0 + row) * D;
        const float* lrow = ldsT + row * D;
        for (int c = threadIdx.x; c < D; c += 256)
            orow[c] = lrow[c] * sc;
    }
}

// ---------------------------------------------------------------------------
extern "C" void kernel_launch(void* const* d_in, const int* in_sizes, int n_in,
                              void* d_out, int out_size, void* d_ws, size_t ws_size,
                              hipStream_t stream) {
    const int B = 32, D = 4096, U = 64;
    const float* stimulus = (const float*)d_in[0];
    const float* prev_act = (const float*)d_in[1];
    const float* W_heb    = (const float*)d_in[2];
    const float* W_rec    = (const float*)d_in[3];
    const float* decay    = (const float*)d_in[4];
    const float* ln_rec_g = (const float*)d_in[5];
    const float* ln_rec_b = (const float*)d_in[6];
    const float* fc1_w    = (const float*)d_in[7];
    const float* fc1_b    = (const float*)d_in[8];
    const float* ln1_g    = (const float*)d_in[9];
    const float* ln1_b    = (const float*)d_in[10];
    const float* fc2_w    = (const float*)d_in[11];
    const float* fc2_b    = (const float*)d_in[12];
    const float* ln2_g    = (const float*)d_in[13];
    const float* ln2_b    = (const float*)d_in[14];
    const float* fc3_w    = (const float*)d_in[15];
    const float* fc3_b    = (const float*)d_in[16];
    const float* lno_g    = (const float*)d_in[17];
    const float* lno_b    = (const float*)d_in[18];
    float* out = (float*)d_out;

    float* ws       = (float*)d_ws;
    float* stim_out = ws;                       // 32*4096
    float* rec_out  = stim_out + (size_t)B * D; // 32*4096
    float* finalb   = rec_out  + (size_t)B * D; // 32*4096
    float* h1       = finalb   + (size_t)B * D; // 32*512
    float* h1n      = h1  + (size_t)B * 512;
    float* h2       = h1n + (size_t)B * 512;    // 32*256
    float* h2n      = h2  + (size_t)B * 256;
    float* h3       = h2n + (size_t)B * 256;    // 32*64
    float* alpha    = h3  + (size_t)B * U;      // 4096

    const dim3 blk(256);

    // both forward GEMMs, one launch: 512 blocks x 8 waves, split-K per block
    fwd_dual_gemm<<<dim3(D / 16, 2), blk, 0, stream>>>(
        stimulus, W_heb, prev_act, W_rec, stim_out, rec_out);

    // final = stim_out + LN(rec_out); also rows 0..31 of d_out
    ln_rec_add<<<dim3(B), blk, 0, stream>>>(rec_out, stim_out, ln_rec_g, ln_rec_b, finalb, out, D);

    // neuromodulator MLP (block per 16-col strip)
    gemm32_block<4096, 512><<<dim3(512 / 16), blk, 0, stream>>>(finalb, fc1_w, fc1_b, h1, 0);
    ln_relu_rows<<<dim3(B), blk, 0, stream>>>(h1, ln1_g, ln1_b, h1n, 512);
    gemm32_block<512, 256><<<dim3(256 / 16), blk, 0, stream>>>(h1n, fc2_w, fc2_b, h2, 0);
    ln_relu_rows<<<dim3(B), blk, 0, stream>>>(h2, ln2_g, ln2_b, h2n, 256);
    gemm32_block<256, 64><<<dim3(U / 16), blk, 0, stream>>>(h2n, fc3_w, fc3_b, h3, 0);
    mlp_tail<<<dim3(1), dim3(1024), 0, stream>>>(h3, lno_g, lno_b, alpha, D);

    // Hebbian weight updates: both matrices, one launch; 16-row panels in
    // 256 KB dynamic LDS
    const size_t ldsBytes = (size_t)16 * D * sizeof(float);
    update_dual<<<dim3(D / 16, 2), blk, ldsBytes, stream>>>(
        W_heb, W_rec, stimulus, prev_act, stim_out, rec_out, alpha, decay,
        out + (size_t)B * D, out + (size_t)(B + D) * D);
}